// ResidualVectorQuantizer_73486890434657
// MI455X (gfx1250) — compile-verified
//
#include <hip/hip_runtime.h>

// ---------------------------------------------------------------------------
// Residual Vector Quantization, fused single kernel for MI455X (gfx1250).
//   z:         [B=16, D=128, T=4096]  f32
//   codebooks: [S=8,  K=1024, D=128]  f32
//   out:       [B, D, T]              f32   (= z - final_residual)
// ~137 GFLOP f32 of GEMM work -> v_wmma_f32_16x16x4_f32 (exact fp32 math so
// argmin indices match the fp32 reference). Residual rows live in registers
// in WMMA A-fragment layout for all 8 stages. Two codeword tiles are
// processed concurrently per wave: two independent WMMA accumulation chains
// interleave (no per-instruction RAW stall) and 8 B-fragment loads stay in
// flight against L2.
// ---------------------------------------------------------------------------

typedef float v2f __attribute__((ext_vector_type(2)));
typedef float v8f __attribute__((ext_vector_type(8)));

#define BB   16
#define DD   128
#define TT   4096
#define NCB  8
#define KK   1024
#define WAVES_PER_BLOCK 8
#define ROWS_PER_WAVE   16
#define BLOCK_T  (WAVES_PER_BLOCK * ROWS_PER_WAVE)   // 128 t-rows per block
#define KSTEPS   (DD / 4)                            // 32 wmma k-steps
#define NTILES   (KK / 16)                           // 64 codeword tiles

__global__ __launch_bounds__(256)
void rvq_fused_kernel(const float* __restrict__ z,
                      const float* __restrict__ cb,
                      float* __restrict__ out) {
    __shared__ float s_cnorm[NCB * KK];                     // 32 KB
    __shared__ int   s_idx[WAVES_PER_BLOCK][ROWS_PER_WAVE]; // winning indices

    const int tid  = threadIdx.x;
    const int w    = tid >> 5;          // wave id in block
    const int lane = tid & 31;
    const int g    = lane >> 4;         // K-half (0: K=0,1 ; 1: K=2,3)
    const int r    = lane & 15;         // row-in-wave / column-in-tile

    const int blk  = blockIdx.x;
    const int b    = blk / (TT / BLOCK_T);
    const int tblk = blk % (TT / BLOCK_T);
    const int t0   = tblk * BLOCK_T + w * ROWS_PER_WAVE;

    // ---- precompute ||c||^2 for all stages into LDS (once per block) ----
    for (int i = tid; i < NCB * KK; i += 256) {
        const float* c = cb + (long)i * DD;
        float s = 0.f;
        #pragma unroll 8
        for (int d = 0; d < DD; ++d) s += c[d] * c[d];
        s_cnorm[i] = s;
    }
    __syncthreads();

    // ---- load this wave's 16 residual rows into registers (A-frag layout) --
    // lane(g,r): areg[kk] = { resid[row=r][4kk+2g], resid[row=r][4kk+2g+1] }
    v2f areg[KSTEPS];
    #pragma unroll
    for (int kk = 0; kk < KSTEPS; ++kk) {
        const int  d0 = 4 * kk + 2 * g;
        const long a0 = ((long)b * DD + d0) * TT + t0 + r;
        areg[kk].x = z[a0];
        areg[kk].y = z[a0 + TT];
    }

    // ---------------------------- stage loop -------------------------------
    for (int s = 0; s < NCB; ++s) {
        const float* cbs = cb + (long)s * KK * DD;

        float best[8];
        int   bidx[8];
        #pragma unroll
        for (int j = 0; j < 8; ++j) { best[j] = 3.4e38f; bidx[j] = 0; }

        // two codeword tiles per iteration: independent WMMA chains
        for (int nt = 0; nt < NTILES; nt += 2) {
            v8f acc0 = {0.f, 0.f, 0.f, 0.f, 0.f, 0.f, 0.f, 0.f};
            v8f acc1 = {0.f, 0.f, 0.f, 0.f, 0.f, 0.f, 0.f, 0.f};
            // B-frag: lane(g, n=r) holds { cb[tile*16+n][d0+2g], [d0+2g+1] }
            const float* cbt0 = cbs + (long)(nt * 16 + r) * DD + 2 * g;
            const float* cbt1 = cbt0 + 16 * DD;
            #pragma unroll
            for (int kk = 0; kk < KSTEPS; ++kk) {
                const v2f bf0 = *(const v2f*)(cbt0 + 4 * kk); // 8B aligned
                const v2f bf1 = *(const v2f*)(cbt1 + 4 * kk);
                acc0 = __builtin_amdgcn_wmma_f32_16x16x4_f32(
                         false, areg[kk], false, bf0, (short)0, acc0, false, false);
                acc1 = __builtin_amdgcn_wmma_f32_16x16x4_f32(
                         false, areg[kk], false, bf1, (short)0, acc1, false, false);
            }
            // score = -2*dot + ||c||^2 ; C layout: slot j, lane -> m=j+8g, n=r
            // merge tile nt first, then nt+1 (strict '<' keeps lowest index)
            const float cn0 = s_cnorm[s * KK + nt * 16 + r];
            const float cn1 = s_cnorm[s * KK + (nt + 1) * 16 + r];
            const int   n0  = nt * 16 + r;
            const int   n1  = (nt + 1) * 16 + r;
            #pragma unroll
            for (int j = 0; j < 8; ++j) {
                const float sc0 = -2.f * acc0[j] + cn0;
                if (sc0 < best[j]) { best[j] = sc0; bidx[j] = n0; }
                const float sc1 = -2.f * acc1[j] + cn1;
                if (sc1 < best[j]) { best[j] = sc1; bidx[j] = n1; }
            }
        }

        // ---- argmin across the 16 columns (butterfly within 16-lane halves)
        #pragma unroll
        for (int m = 1; m < 16; m <<= 1) {
            #pragma unroll
            for (int j = 0; j < 8; ++j) {
                const float os = __shfl_xor(best[j], m, 32);
                const int   oi = __shfl_xor(bidx[j], m, 32);
                if (os < best[j] || (os == best[j] && oi < bidx[j])) {
                    best[j] = os; bidx[j] = oi;
                }
            }
        }
        // slot j in half g is the winner for row m = j + 8*g
        if (r == 0) {
            #pragma unroll
            for (int j = 0; j < 8; ++j) s_idx[w][j + 8 * g] = bidx[j];
        }
        __syncthreads();
        const int myIdx = s_idx[w][r];   // index for this lane's row r

        // ---- residual -= chosen codeword (registers, float2 slices) -------
        const float* q = cbs + (long)myIdx * DD + 2 * g;
        #pragma unroll
        for (int kk = 0; kk < KSTEPS; ++kk) {
            const v2f qv = *(const v2f*)(q + 4 * kk);
            areg[kk].x -= qv.x;
            areg[kk].y -= qv.y;
        }
        __syncthreads();  // protect s_idx reuse next stage (uniform flow)
    }

    // ---- out = z - final residual  (== sum of selected codewords) ---------
    #pragma unroll
    for (int kk = 0; kk < KSTEPS; ++kk) {
        const int  d0 = 4 * kk + 2 * g;
        const long a0 = ((long)b * DD + d0) * TT + t0 + r;
        out[a0]      = z[a0]      - areg[kk].x;
        out[a0 + TT] = z[a0 + TT] - areg[kk].y;
    }
}

extern "C" void kernel_launch(void* const* d_in, const int* in_sizes, int n_in,
                              void* d_out, int out_size, void* d_ws, size_t ws_size,
                              hipStream_t stream) {
    (void)in_sizes; (void)n_in; (void)d_ws; (void)ws_size; (void)out_size;
    const float* z  = (const float*)d_in[0];   // [16,128,4096] f32
    const float* cb = (const float*)d_in[1];   // [8,1024,128]  f32
    float* out      = (float*)d_out;           // [16,128,4096] f32

    const int grid = BB * (TT / BLOCK_T);      // 16 * 32 = 512 blocks
    rvq_fused_kernel<<<grid, 256, 0, stream>>>(z, cb, out);
}